// LowpassCell_72980084294306
// MI455X (gfx1250) — compile-verified
//
#include <hip/hip_runtime.h>
#include <cstdint>
#include <cstddef>

// ---------------- problem constants (from reference) ----------------
constexpr float kDT = 0.001f;
constexpr int Bn = 32;          // batch
constexpr int Tn = 4096;        // timesteps
constexpr int Nn = 128;         // features
constexpr int Ln = 64;          // timesteps per chunk
constexpr int Cn = Tn / Ln;     // 64 chunks
constexpr int STAGE = 16;       // timesteps per async-prefetch stage
constexpr int NST = Ln / STAGE; // 4 stages per chunk
constexpr int WPB = 4;          // waves per block (wave32)
constexpr int BLOCK = WPB * 32;

// ---------------- vector types (clang ext vectors, NT-store capable) --------
typedef float v4f __attribute__((ext_vector_type(4)));
typedef int   v4i __attribute__((ext_vector_type(4)));
typedef __attribute__((address_space(1))) v4i* gas_v4i;
typedef __attribute__((address_space(3))) v4i* lds_v4i;

// ---------------- CDNA5 async global->LDS copy (ASYNCcnt path) ----------------
__device__ __forceinline__ void async_copy_b128(const void* g, void* l) {
#if __has_builtin(__builtin_amdgcn_global_load_async_to_lds_b128)
    // (global v4i32* src, LDS v4i32* dst, imm offset, imm cpol)
    __builtin_amdgcn_global_load_async_to_lds_b128((gas_v4i)g, (lds_v4i)l, 0, 0);
#else
    uint64_t ga = (uint64_t)(uintptr_t)g;
    uint32_t la = (uint32_t)(uintptr_t)l;   // low 32 bits of generic shared ptr = LDS offset
    asm volatile("global_load_async_to_lds_b128 %0, %1, off"
                 :: "v"(la), "v"(ga) : "memory");
#endif
}

// s_wait_asynccnt as raw asm: also acts as a compiler memory barrier so the
// LDS reads can't be hoisted above the wait.
#define WAIT_ASYNCCNT(n) asm volatile("s_wait_asynccnt %0" :: "i"(n) : "memory")

// ---------------- pass 1 / pass 3 kernel ----------------
// One wave per (batch b, chunk c). Lanes cover N=128 as 32 x float4.
// FINAL=false: compute per-chunk local sum S into carry[].
// FINAL=true : read carry-in from carry[], write outputs (non-temporal).
template <bool FINAL>
__global__ __launch_bounds__(BLOCK)
void lowpass_chunk(const float* __restrict__ in,
                   const float* __restrict__ tau,
                   float* __restrict__ carry,
                   float* __restrict__ out) {
    __shared__ v4f smem[WPB][2][STAGE * 32];   // 64 KB static LDS

    const int lane = threadIdx.x & 31;
    const int w    = threadIdx.x >> 5;
    const int p    = blockIdx.x * WPB + w;     // chunk-row id: 0 .. Bn*Cn-1
    const int b    = p / Cn;
    const int c    = p % Cn;
    const int n0   = lane * 4;

    // per-channel decay a = exp(-dt / max(tau,1e-8)) and gain (1-a)
    const v4f t4 = *(const v4f*)(tau + n0);
    v4f A, G;
    A.x = expf(-kDT / fmaxf(t4.x, 1e-8f));
    A.y = expf(-kDT / fmaxf(t4.y, 1e-8f));
    A.z = expf(-kDT / fmaxf(t4.z, 1e-8f));
    A.w = expf(-kDT / fmaxf(t4.w, 1e-8f));
    G.x = 1.0f - A.x; G.y = 1.0f - A.y; G.z = 1.0f - A.z; G.w = 1.0f - A.w;

    v4f x;
    if (FINAL) {
        x = *(const v4f*)(carry + (size_t)p * Nn + n0);   // carry-in x_{cL-1}
    } else {
        x.x = 0.0f; x.y = 0.0f; x.z = 0.0f; x.w = 0.0f;
    }

    const float* gbase = in + ((size_t)b * Tn + (size_t)c * Ln) * Nn;
    float* obase = FINAL ? out + ((size_t)b * Tn + (size_t)c * Ln) * Nn : nullptr;

    // prefetch stage 0
    {
        const float* g = gbase + n0;
#pragma unroll
        for (int jj = 0; jj < STAGE; ++jj)
            async_copy_b128(g + (size_t)jj * Nn, &smem[w][0][jj * 32 + lane]);
    }

#pragma unroll
    for (int s = 0; s < NST; ++s) {
        if (s + 1 < NST) {
            const float* g = gbase + (size_t)(s + 1) * STAGE * Nn + n0;
#pragma unroll
            for (int jj = 0; jj < STAGE; ++jj)
                async_copy_b128(g + (size_t)jj * Nn, &smem[w][(s + 1) & 1][jj * 32 + lane]);
            WAIT_ASYNCCNT(STAGE);   // stage s landed; stage s+1 still in flight
        } else {
            WAIT_ASYNCCNT(0);
        }

        const v4f* buf = smem[w][s & 1];
#pragma unroll
        for (int jj = 0; jj < STAGE; ++jj) {
            v4f u = buf[jj * 32 + lane];
            x.x = fmaf(A.x, x.x, G.x * u.x);
            x.y = fmaf(A.y, x.y, G.y * u.y);
            x.z = fmaf(A.z, x.z, G.z * u.z);
            x.w = fmaf(A.w, x.w, G.w * u.w);
            if (FINAL) {
                __builtin_nontemporal_store(
                    x, (v4f*)(obase + (size_t)(s * STAGE + jj) * Nn + n0));
            }
        }
    }

    if (!FINAL)
        *(v4f*)(carry + (size_t)p * Nn + n0) = x;   // S_c (chunk local sum)
}

// ---------------- pass 2: scan the chunk carries ----------------
// One thread per (b, n-quad). In-place: carry[] holds S_c on entry,
// carry-in x_{cL-1} on exit.  y_{c+1} = a^L * y_c + S_c,  y_0 = initial_level.
__global__ void lowpass_carry_scan(const float* __restrict__ tau,
                                   const float* __restrict__ x0,
                                   float* __restrict__ carry) {
    const int tid = blockIdx.x * blockDim.x + threadIdx.x;
    if (tid >= Bn * (Nn / 4)) return;
    const int b  = tid / (Nn / 4);
    const int n0 = (tid % (Nn / 4)) * 4;

    const v4f t4 = *(const v4f*)(tau + n0);
    v4f AL;
    AL.x = expf(-((float)Ln * kDT) / fmaxf(t4.x, 1e-8f));
    AL.y = expf(-((float)Ln * kDT) / fmaxf(t4.y, 1e-8f));
    AL.z = expf(-((float)Ln * kDT) / fmaxf(t4.z, 1e-8f));
    AL.w = expf(-((float)Ln * kDT) / fmaxf(t4.w, 1e-8f));

    v4f y = *(const v4f*)(x0 + n0);   // initial_level broadcast over batch
    for (int c = 0; c < Cn; ++c) {
        v4f* ptr = (v4f*)(carry + ((size_t)b * Cn + c) * Nn + n0);
        v4f S = *ptr;
        *ptr = y;                            // carry-in for chunk c
        y.x = fmaf(AL.x, y.x, S.x);
        y.y = fmaf(AL.y, y.y, S.y);
        y.z = fmaf(AL.z, y.z, S.z);
        y.w = fmaf(AL.w, y.w, S.w);
    }
}

// ---------------- launcher ----------------
extern "C" void kernel_launch(void* const* d_in, const int* in_sizes, int n_in,
                              void* d_out, int out_size, void* d_ws, size_t ws_size,
                              hipStream_t stream) {
    (void)in_sizes; (void)n_in; (void)out_size; (void)ws_size;
    const float* inputs = (const float*)d_in[0];   // [B, T, N] f32
    const float* init   = (const float*)d_in[1];   // [1, N]   f32
    const float* tau    = (const float*)d_in[2];   // [1, N]   f32
    float* out   = (float*)d_out;                  // [B, T, N] f32
    float* carry = (float*)d_ws;                   // B*C*N floats = 1 MB scratch

    const int rows = Bn * Cn;                      // 2048 chunk-rows (one wave each)
    lowpass_chunk<false><<<rows / WPB, BLOCK, 0, stream>>>(inputs, tau, carry, out);

    const int t2 = Bn * (Nn / 4);                  // 1024 threads
    lowpass_carry_scan<<<(t2 + 255) / 256, 256, 0, stream>>>(tau, init, carry);

    lowpass_chunk<true><<<rows / WPB, BLOCK, 0, stream>>>(inputs, tau, carry, out);
}